// Anchor3DHead_47064251629653
// MI455X (gfx1250) — compile-verified
//
#include <hip/hip_runtime.h>
#include <cstdint>
#include <cstddef>

// ---------------------------------------------------------------------------
// Anchor3DHead fused 1x1-conv heads for MI455X (gfx1250).
// Memory-bound: 329MB of fp32 activations @ 23.3TB/s ~= 15us floor.
// Strategy: stream x exactly once via async global->LDS copies (ASYNCcnt,
// double-buffered), contract K=384 channels for all 3 heads with full-
// precision V_WMMA_F32_16X16X4_F32 (no bf16 conversion VALU, bit-accurate
// fp32 math vs the reference).
// ---------------------------------------------------------------------------

typedef __attribute__((ext_vector_type(2))) float v2f;
typedef __attribute__((ext_vector_type(8))) float v8f;

constexpr int kB    = 4;
constexpr int kCin  = 384;
constexpr int kH    = 248;
constexpr int kW    = 216;
constexpr int kS    = kH * kW;                 // 53568
constexpr int kKC   = 32;                      // K rows staged per chunk
constexpr int kNCH  = kCin / kKC;              // 12 chunks
constexpr int kTS   = 128;                     // spatial cols per block (8 waves * 16)
constexpr int kTiles = (kS + kTS - 1) / kTS;   // 419
constexpr int kXP   = 132;                     // padded X-tile row stride (floats)
constexpr int kWP   = 388;                     // padded weight row stride (384%64==0 -> pad)

// Concatenated output rows: 0..1 = cls, 2..15 = reg, 16..19 = dir, 20..31 = zero pad
__device__ __forceinline__ float wval(const float* __restrict__ wc,
                                      const float* __restrict__ wr,
                                      const float* __restrict__ wd,
                                      int c, int m) {
    if (m < 2)  return wc[c * 2  + m];
    if (m < 16) return wr[c * 14 + (m - 2)];
    if (m < 20) return wd[c * 4  + (m - 16)];
    return 0.0f;
}
__device__ __forceinline__ float biasv(const float* __restrict__ bc,
                                       const float* __restrict__ br,
                                       const float* __restrict__ bd, int m) {
    if (m < 2)  return bc[m];
    if (m < 16) return br[m - 2];
    return bd[m - 16];
}
// d_out = [cls 4x2xS | reg 4x14xS | dir 4x4xS] flat
__device__ __forceinline__ size_t outoff(int b, int m, int s) {
    if (m < 2)  return ((size_t)(b * 2  +  m      )) * kS + s;
    if (m < 16) return (size_t)8  * kS + ((size_t)(b * 14 + (m - 2 ))) * kS + s;
    return       (size_t)64 * kS + ((size_t)(b * 4  + (m - 16))) * kS + s;
}

__global__ __launch_bounds__(256)
void anchor3dhead_fused(const float* __restrict__ x,
                        const float* __restrict__ wcls, const float* __restrict__ bcls,
                        const float* __restrict__ wreg, const float* __restrict__ breg,
                        const float* __restrict__ wdir, const float* __restrict__ bdir,
                        float* __restrict__ out)
{
    __shared__ float Wlds[32][kWP];            // 48.5 KB  fused weight matrix W[m][c]
    __shared__ float Xlds[2][kKC][kXP];        // 2x16.5KB double-buffered x tile [k][s]

    const int t    = threadIdx.x;
    const int lane = t & 31;
    const int wv   = t >> 5;                   // wave id 0..7 -> 16-col strip
    const int half = lane >> 4;                // lane half selects K (A/B) or M+8 (C/D)
    const int n    = lane & 15;
    const int b    = blockIdx.y;
    const int s0   = blockIdx.x * kTS;
    const float* xb = x + (size_t)b * kCin * kS;

    // ---- fused weights (fp32, zero-padded to 32 rows) into LDS, once ----
    for (int idx = t; idx < 32 * kCin; idx += 256) {
        int m = idx / kCin;
        int c = idx - m * kCin;
        Wlds[m][c] = wval(wcls, wreg, wdir, c, m);
    }

    // ---- async copy of one K-chunk (32 x 128 floats) into an LDS buffer ----
    auto issue_chunk = [&](int ch, int buf) {
        uint32_t lbase = (uint32_t)(uintptr_t)&Xlds[buf][0][0];
        #pragma unroll
        for (int i = 0; i < 4; ++i) {           // 1024 b128 moves / 256 threads
            int e   = t + 256 * i;
            int row = e >> 5;                   // K row 0..31
            int c16 = e & 31;                   // 16-byte column 0..31
            int s   = s0 + c16 * 4;
            if (s > kS - 4) s = kS - 4;         // clamp: EXEC stays full, 16B aligned
            const float* g = xb + (size_t)(ch * kKC + row) * kS + s;
            uint32_t loff = lbase + (uint32_t)row * (kXP * 4) + (uint32_t)c16 * 16;
            asm volatile("global_load_async_to_lds_b128 %0, %1, off"
                         :: "v"(loff), "v"(g) : "memory");
        }
    };

    v8f acc0 = {};   // output rows 0..15  (VGPR r -> M = r + 8*half)
    v8f acc1 = {};   // output rows 16..31 (16..19 valid, rest multiply by zero pad)
    const int colb = wv * 16 + n;

    issue_chunk(0, 0);
    for (int ch = 0; ch < kNCH; ++ch) {
        const int buf = ch & 1;
        if (ch + 1 < kNCH) {
            issue_chunk(ch + 1, buf ^ 1);                 // prefetch next chunk
            asm volatile("s_wait_asynccnt 0x4" ::: "memory");  // chunk ch landed
        } else {
            asm volatile("s_wait_asynccnt 0x0" ::: "memory");
        }
        __syncthreads();   // all waves' async data visible (also covers Wlds fill)

        #pragma unroll
        for (int kk = 0; kk < 8; ++kk) {       // 8 x (K=4) steps per chunk
            const int kloc  = kk * 4 + 2 * half;
            const int kglob = ch * kKC + kloc;
            // A fragments (16x4 f32): VGPR0 = K(2*half), VGPR1 = K(2*half+1)
            v2f a0 = *(const v2f*)&Wlds[n][kglob];
            v2f a1 = *(const v2f*)&Wlds[16 + n][kglob];
            // B fragment (4x16 f32): two consecutive K rows at this lane's column
            v2f bv = { Xlds[buf][kloc][colb], Xlds[buf][kloc + 1][colb] };
            acc0 = __builtin_amdgcn_wmma_f32_16x16x4_f32(false, a0, false, bv,
                                                         (short)0, acc0, false, false);
            acc1 = __builtin_amdgcn_wmma_f32_16x16x4_f32(false, a1, false, bv,
                                                         (short)0, acc1, false, false);
        }
        __syncthreads();   // release this buffer for the next async overwrite
    }

    // ---- add bias, scatter the 20 valid output rows (coalesced along s) ----
    const int s = s0 + colb;
    if (s < kS) {
        #pragma unroll
        for (int r = 0; r < 8; ++r) {
            int m = r + 8 * half;                          // 0..15
            out[outoff(b, m, s)] = acc0[r] + biasv(bcls, breg, bdir, m);
        }
        if (half == 0) {
            #pragma unroll
            for (int r = 0; r < 4; ++r) {
                int m = 16 + r;                            // 16..19 (dir head)
                out[outoff(b, m, s)] = acc1[r] + biasv(bcls, breg, bdir, m);
            }
        }
    }
}

extern "C" void kernel_launch(void* const* d_in, const int* in_sizes, int n_in,
                              void* d_out, int out_size, void* d_ws, size_t ws_size,
                              hipStream_t stream) {
    (void)in_sizes; (void)n_in; (void)out_size; (void)d_ws; (void)ws_size;
    const float* x    = (const float*)d_in[0];
    const float* wcls = (const float*)d_in[1];
    const float* bcls = (const float*)d_in[2];
    const float* wreg = (const float*)d_in[3];
    const float* breg = (const float*)d_in[4];
    const float* wdir = (const float*)d_in[5];
    const float* bdir = (const float*)d_in[6];
    float* out = (float*)d_out;

    dim3 grid(kTiles, kB);   // 419 spatial tiles x 4 batches, 8 waves per block
    anchor3dhead_fused<<<grid, 256, 0, stream>>>(x, wcls, bcls, wreg, breg,
                                                 wdir, bdir, out);
}